// KappaGCN_70291434766337
// MI455X (gfx1250) — compile-verified
//
#include <hip/hip_runtime.h>
#include <hip/hip_bf16.h>
#include <math.h>
#include <stdint.h>

// ---------------------------------------------------------------------------
// KappaGCN forward for MI455X (gfx1250, wave32, WMMA + async LDS copies).
//   N=8192, D=256, K=16, kappa=-1 (Poincare ball, SQ=1)
// Pipeline:
//   K0: per-class constants
//   K1: Mx = X @ W                       (WMMA f32 GEMM)
//   K2: rowwise mobius_matvec tail -> Sext[n] = [lam*XW | lam-1 | 1 | 0...]
//   K3: NumExt = A_hat @ Sext            (dominant GEMM, one pass over 256MB)
//   K4: rowwise midpoint/scalar-mul/log-relu-exp + hyperbolic MLR logits
//   K5: OutPad = A_hat @ LogitsPad
//   K6: strip padding -> d_out [N,K]
// GEMM: 128x32 block tile, 8 waves (4 row-groups x 2 col-groups), each wave
// two 16x16 f32 accumulators; K panels of 32 double-buffered in LDS, filled
// by global_load_async_to_lds_b128 (ASYNCcnt) overlapped with WMMA compute.
// ---------------------------------------------------------------------------

#define NN   8192
#define DD   256
#define KK   16
#define DEXT 288          // 256 data cols + col256 = lam-1, col257 = 1, pad
#define LPAD 32
#define EPSF 1e-10f

#define TM   128          // block tile rows
#define TN   32           // block tile cols
#define KB   32           // K panel
#define LSTR 44           // LDS row stride (floats): 16B aligned, bank-friendly

typedef float v2f __attribute__((ext_vector_type(2)));
typedef float v8f __attribute__((ext_vector_type(8)));

__device__ __forceinline__ uint32_t lds_addr_u32(const void* p) {
    // addrspace(3) -> generic keeps the LDS offset in the low 32 bits
    return (uint32_t)(uintptr_t)p;
}

// CDNA5 async memory->LDS copy, 16B per lane, tracked by ASYNCcnt.
__device__ __forceinline__ void async_copy_b128(uint32_t lds_byte,
                                                const float* gptr) {
    asm volatile("global_load_async_to_lds_b128 %0, %1, off"
                 :: "v"(lds_byte), "v"(gptr)
                 : "memory");
}

__device__ __forceinline__ void wait_async0() {
    asm volatile("s_wait_asynccnt 0x0" ::: "memory");
}

__device__ __forceinline__ float atanh_c(float v) {
    v = fminf(fmaxf(v, -1.0f + 1e-7f), 1.0f - 1e-7f);
    return atanhf(v);
}

__device__ __forceinline__ float block_reduce_sum(float v, float* sh) {
    const int tid = threadIdx.x;
    sh[tid] = v;
    __syncthreads();
    #pragma unroll
    for (int s = 128; s > 0; s >>= 1) {
        if (tid < s) sh[tid] += sh[tid + s];
        __syncthreads();
    }
    float r = sh[0];
    __syncthreads();
    return r;
}

// ---------------------------------------------------------------------------
// f32 WMMA GEMM: C[M,N] = A[M,K] * B[K,N], row-major.
// Requires M%128==0, N%32==0, K%32==0, lda/ldb multiples of 4 (b128 align).
// ---------------------------------------------------------------------------
__global__ __launch_bounds__(256)
void gemm_f32_wmma(const float* __restrict__ A, const float* __restrict__ B,
                   float* __restrict__ C, int M, int N, int K,
                   int lda, int ldb, int ldc) {
    __shared__ float As[2][TM * LSTR];   // 2 x 128x32 panel (stride 44)
    __shared__ float Bs[2][KB * LSTR];   // 2 x  32x32 panel

    const int tid  = threadIdx.x;
    const int lane = tid & 31;
    const int wave = tid >> 5;
    const int rt   = wave & 3;           // row group: 32 rows each
    const int ct   = wave >> 2;          // col group: 16 cols each
    const int row0 = blockIdx.x * TM;
    const int col0 = blockIdx.y * TN;

    const int m16 = lane & 15;           // M(A)/N(B,C,D) index within tile
    const int ko  = (lane >> 4) << 1;    // K sub-offset: 0 or 2
    const int mhi = (lane >> 4) << 3;    // C/D: +8 rows for upper half-wave

    // per-thread async-copy coordinates
    const int ar = tid >> 3;             // A: base row (0..31 -> x4 groups)
    const int ac = (tid & 7) << 2;       // A: col (float4 granularity)
    const int br = tid >> 3;             // B: k row 0..31
    const int bc = (tid & 7) << 2;       // B: col

    const uint32_t asb[2] = { lds_addr_u32(&As[0][0]), lds_addr_u32(&As[1][0]) };
    const uint32_t bsb[2] = { lds_addr_u32(&Bs[0][0]), lds_addr_u32(&Bs[1][0]) };

    v8f acc0 = {};
    v8f acc1 = {};

    // issue async panel copies for K-panel at k0 into buffer p
    auto issue = [&](int k0, int p) {
        #pragma unroll
        for (int t = 0; t < 4; ++t) {
            const int r = ar + t * 32;   // 0..127
            async_copy_b128(asb[p] + (uint32_t)(r * LSTR + ac) * 4u,
                            A + (size_t)(row0 + r) * lda + (k0 + ac));
        }
        async_copy_b128(bsb[p] + (uint32_t)(br * LSTR + bc) * 4u,
                        B + (size_t)(k0 + br) * ldb + (col0 + bc));
    };

    // prologue: fill buffer 0
    issue(0, 0);
    wait_async0();
    __syncthreads();

    int p = 0;
    for (int k0 = 0; k0 < K; k0 += KB) {
        if (k0 + KB < K) issue(k0 + KB, p ^ 1);   // overlap next panel

        const float* Asp = &As[p][0];
        const float* Bsp = &Bs[p][0];
        const int arow0 = (rt * 32 + m16) * LSTR;
        const int arow1 = (rt * 32 + 16 + m16) * LSTR;
        const int bcol  = ct * 16 + m16;

        #pragma unroll
        for (int kb = 0; kb < KB; kb += 4) {
            v2f a0, a1, b;
            a0.x = Asp[arow0 + kb + ko];
            a0.y = Asp[arow0 + kb + ko + 1];
            a1.x = Asp[arow1 + kb + ko];
            a1.y = Asp[arow1 + kb + ko + 1];
            b.x  = Bsp[(kb + ko) * LSTR + bcol];
            b.y  = Bsp[(kb + ko + 1) * LSTR + bcol];
            acc0 = __builtin_amdgcn_wmma_f32_16x16x4_f32(
                false, a0, false, b, (short)0, acc0, false, false);
            acc1 = __builtin_amdgcn_wmma_f32_16x16x4_f32(
                false, a1, false, b, (short)0, acc1, false, false);
        }
        wait_async0();       // own async copies into p^1 are done
        __syncthreads();     // everyone done reading p / writing p^1
        p ^= 1;
    }

    // C/D layout: N = lane%16, M = vgpr + 8*(lane/16)
    const int nOut = col0 + ct * 16 + m16;
    const int mB0  = row0 + rt * 32 + mhi;
    const int mB1  = mB0 + 16;
    #pragma unroll
    for (int v = 0; v < 8; ++v) {
        C[(size_t)(mB0 + v) * ldc + nOut] = acc0[v];
    }
    #pragma unroll
    for (int v = 0; v < 8; ++v) {
        C[(size_t)(mB1 + v) * ldc + nOut] = acc1[v];
    }
}

// ---------------------------------------------------------------------------
// K0: per-class constants for hyperbolic MLR.
// ---------------------------------------------------------------------------
__global__ void class_consts_kernel(const float* __restrict__ pks,
                                    const float* __restrict__ Wl,
                                    float* __restrict__ consts) {
    const int k = threadIdx.x;
    if (k < KK) {
        float b2 = 0.f, an2 = 0.f, bW = 0.f;
        for (int d = 0; d < DD; ++d) {
            float b = pks[k * DD + d];
            float w = Wl[d * KK + k];
            b2 += b * b; an2 += w * w; bW += b * w;
        }
        consts[k * 4 + 0] = b2;
        consts[k * 4 + 1] = fmaxf(sqrtf(an2), EPSF);
        consts[k * 4 + 2] = bW;
        consts[k * 4 + 3] = 2.0f / (1.0f - b2);     // kappa = -1
    }
}

// ---------------------------------------------------------------------------
// K2: mobius_matvec tail + conformal factor -> padded B for big GEMM.
// ---------------------------------------------------------------------------
__global__ __launch_bounds__(256)
void rowwise_xw_lam_kernel(const float* __restrict__ X,
                           const float* __restrict__ Mx,
                           float* __restrict__ Sext) {
    __shared__ float sh[256];
    const int n = blockIdx.x, d = threadIdx.x;
    const float x  = X[(size_t)n * DD + d];
    const float mx = Mx[(size_t)n * DD + d];
    const float xn2  = block_reduce_sum(x * x, sh);
    const float mxn2 = block_reduce_sum(mx * mx, sh);
    const float xn  = fmaxf(sqrtf(xn2), EPSF);
    const float mxn = fmaxf(sqrtf(mxn2), EPSF);
    const float s   = tanhf((mxn / xn) * atanh_c(xn));  // = ||XW||
    const float xw  = (s / mxn) * mx;
    const float lam = 2.0f / (1.0f - s * s);
    Sext[(size_t)n * DEXT + d] = lam * xw;
    if (d < 32) {
        float v = 0.0f;
        if (d == 0) v = lam - 1.0f;       // den column
        else if (d == 1) v = 1.0f;        // rowsum(A) column
        Sext[(size_t)n * DEXT + DD + d] = v;
    }
}

// ---------------------------------------------------------------------------
// K4: weighted midpoint -> mobius scalar muls -> exp(relu(log)) -> MLR logits
// ---------------------------------------------------------------------------
__global__ __launch_bounds__(256)
void rowwise_logits_kernel(const float* __restrict__ NumExt,
                           const float* __restrict__ pks,
                           const float* __restrict__ Wl,
                           const float* __restrict__ consts,
                           float* __restrict__ LogitsPad) {
    __shared__ float sh[256];
    __shared__ float h1[256];
    const int n = blockIdx.x, d = threadIdx.x;
    const float* row = NumExt + (size_t)n * DEXT;

    float den  = row[DD];
    float rsum = row[DD + 1];
    den = (fabsf(den) < EPSF) ? EPSF : den;

    const float y   = row[d] / den;
    const float ny2 = block_reduce_sum(y * y, sh);
    const float ny  = fmaxf(sqrtf(ny2), EPSF);
    const float t1  = tanhf(0.5f * atanh_c(ny));            // ||mid||
    const float mid = (t1 / ny) * y;
    const float nm  = fmaxf(t1, EPSF);
    const float t2  = tanhf(rsum * atanh_c(nm));            // ||h||
    const float h   = (t2 / nm) * mid;
    const float nh  = fmaxf(fabsf(t2), EPSF);
    const float u   = (atanh_c(nh) / nh) * h;               // logmap0
    const float ru  = fmaxf(u, 0.0f);                       // relu
    const float un2 = block_reduce_sum(ru * ru, sh);
    const float un  = fmaxf(sqrtf(un2), EPSF);
    const float th  = tanhf(un);
    const float H1  = (th / un) * ru;                       // expmap0
    const float y2H = th * th;                              // ||H1||^2

    h1[d] = H1;
    __syncthreads();

    if (d < LPAD) {
        float out = 0.0f;
        if (d < KK) {
            const int k = d;
            float dk = 0.f, xw = 0.f;
            for (int j = 0; j < DD; ++j) {
                const float hv = h1[j];
                dk += hv * pks[k * DD + j];
                xw += hv * Wl[j * KK + k];
            }
            const float b2   = consts[k * 4 + 0];
            const float an   = consts[k * 4 + 1];
            const float bW   = consts[k * 4 + 2];
            const float lamb = consts[k * 4 + 3];
            const float c1 = 1.0f - 2.0f * dk + y2H;
            const float c2 = 1.0f - b2;
            float dm = 1.0f - 2.0f * dk + b2 * y2H;
            dm = fmaxf(dm, 1e-15f);
            float zn2 = (c1 * c1 * b2 - 2.0f * c1 * c2 * dk + c2 * c2 * y2H)
                        / (dm * dm);
            const float zn = fmaxf(sqrtf(fmaxf(zn2, 0.0f)), EPSF);
            const float za = (-c1 * bW + c2 * xw) / dm;
            float dist = 2.0f * za / ((1.0f - zn * zn) * an);
            dist = asinhf(dist);
            out = lamb * an * dist;
        }
        LogitsPad[(size_t)n * LPAD + d] = out;
    }
}

// K6: strip padding
__global__ __launch_bounds__(256)
void copy_out_kernel(const float* __restrict__ OutPad, float* __restrict__ out) {
    const int i = blockIdx.x * 256 + threadIdx.x;
    if (i < NN * KK) out[i] = OutPad[(size_t)(i / KK) * LPAD + (i % KK)];
}

// ---------------------------------------------------------------------------
extern "C" void kernel_launch(void* const* d_in, const int* in_sizes, int n_in,
                              void* d_out, int out_size, void* d_ws, size_t ws_size,
                              hipStream_t stream) {
    const float* X    = (const float*)d_in[0];   // [N,D]
    const float* Ah   = (const float*)d_in[1];   // [N,N]
    const float* W    = (const float*)d_in[2];   // [D,D]
    const float* Wl   = (const float*)d_in[3];   // [D,K]
    const float* pks  = (const float*)d_in[4];   // [K,D]
    float* out = (float*)d_out;                  // [N,K]

    float* ws = (float*)d_ws;
    float* Mx        = ws;                                   // N*D
    float* Sext      = Mx        + (size_t)NN * DD;          // N*DEXT
    float* NumExt    = Sext      + (size_t)NN * DEXT;        // N*DEXT
    float* LogitsPad = NumExt    + (size_t)NN * DEXT;        // N*LPAD
    float* OutPad    = LogitsPad + (size_t)NN * LPAD;        // N*LPAD
    float* consts    = OutPad    + (size_t)NN * LPAD;        // 64

    // K0: class constants
    class_consts_kernel<<<1, 32, 0, stream>>>(pks, Wl, consts);

    // K1: Mx = X @ W   (8192 x 256 x 256)
    gemm_f32_wmma<<<dim3(NN / TM, DD / TN), 256, 0, stream>>>(
        X, W, Mx, NN, DD, DD, DD, DD, DD);

    // K2: mobius_matvec tail + lambda, build padded B
    rowwise_xw_lam_kernel<<<NN, 256, 0, stream>>>(X, Mx, Sext);

    // K3: NumExt = A_hat @ Sext   (8192 x 288 x 8192) -- dominant GEMM
    gemm_f32_wmma<<<dim3(NN / TM, DEXT / TN), 256, 0, stream>>>(
        Ah, Sext, NumExt, NN, DEXT, NN, NN, DEXT, DEXT);

    // K4: midpoint + activation + hyperbolic MLR logits
    rowwise_logits_kernel<<<NN, 256, 0, stream>>>(
        NumExt, pks, Wl, consts, LogitsPad);

    // K5: OutPad = A_hat @ LogitsPad   (8192 x 32 x 8192)
    gemm_f32_wmma<<<dim3(NN / TM, LPAD / TN), 256, 0, stream>>>(
        Ah, LogitsPad, OutPad, NN, LPAD, NN, NN, LPAD, LPAD);

    // K6: strip padding into d_out
    copy_out_kernel<<<(NN * KK + 255) / 256, 256, 0, stream>>>(OutPad, out);
}